// MapNetWithIMUCriterion_34591666602402
// MI455X (gfx1250) — compile-verified
//
#include <hip/hip_runtime.h>
#include <hip/hip_bf16.h>
#include <math.h>

#define EPSF 1e-8f
#define TT   1024
#define NN   2048
#define NTHR 256

typedef __attribute__((ext_vector_type(2))) float v2f;
typedef __attribute__((ext_vector_type(8))) float v8f;

struct Quat { float w, x, y, z; };

__device__ __forceinline__ Quat qmul(const Quat a, const Quat b) {
    Quat r;
    r.w = a.w*b.w - a.x*b.x - a.y*b.y - a.z*b.z;
    r.x = a.w*b.x + a.x*b.w + a.y*b.z - a.z*b.y;
    r.y = a.w*b.y - a.x*b.z + a.y*b.w + a.z*b.x;
    r.z = a.w*b.z + a.x*b.y - a.y*b.x + a.z*b.w;
    return r;
}

__device__ __forceinline__ Quat qexp3(float vx, float vy, float vz) {
    float n = fmaxf(sqrtf(vx*vx + vy*vy + vz*vz), EPSF);
    float s = sinf(n) / n;
    Quat q; q.w = cosf(n); q.x = s*vx; q.y = s*vy; q.z = s*vz;
    return q;
}

__device__ __forceinline__ void qlog3(const Quat q, float& ox, float& oy, float& oz) {
    float n = fmaxf(sqrtf(q.x*q.x + q.y*q.y + q.z*q.z), EPSF);
    float a = acosf(fminf(fmaxf(q.w, -1.0f), 1.0f)) / n;
    ox = q.x*a; oy = q.y*a; oz = q.z*a;
}

// One block per trajectory n; 256 threads grid-stride over t.
// Produces 6 partial sums per block into ws[blockIdx*6 + j]:
//   0: |pred.xyz - targ.xyz|          (all T)
//   1: |pred.q   - targ.q  |          (all T)
//   2: |pred_vos.xyz - targ_vos.xyz|  (T-1)
//   3: |pred_vos.q   - targ_vos.q  |  (T-1)
//   4: |pred_vos.xyz - imu_trans   |  (T-1)
//   5: |targ_imu_q   - imu_ori_shift| (T-1)
__global__ __launch_bounds__(NTHR) void mapnet_partial(
        const float* __restrict__ pred, const float* __restrict__ targ,
        const float* __restrict__ imu, float* __restrict__ ws) {
    const int n   = blockIdx.x;
    const int tid = threadIdx.x;
    const long long base = (long long)n * TT;

    float s0 = 0.f, s1 = 0.f, s2 = 0.f, s3 = 0.f, s4 = 0.f, s5 = 0.f;

    for (int t = tid; t < TT; t += NTHR) {
        const float* pr = pred + (base + t) * 6;
        const float* tr = targ + (base + t) * 6;
        float p0 = pr[0], p1 = pr[1], p2 = pr[2], p3 = pr[3], p4 = pr[4], p5 = pr[5];
        float g0 = tr[0], g1 = tr[1], g2 = tr[2], g3 = tr[3], g4 = tr[4], g5 = tr[5];

        s0 += fabsf(p0-g0) + fabsf(p1-g1) + fabsf(p2-g2);
        s1 += fabsf(p3-g3) + fabsf(p4-g4) + fabsf(p5-g5);

        if (t < TT - 1) {
            const float* prn = pr + 6;
            const float* trn = tr + 6;
            float h0 = trn[0], h1 = trn[1], h2 = trn[2], h3 = trn[3], h4 = trn[4], h5 = trn[5];
            // velocity-odometry diffs
            float pv0 = prn[0]-p0, pv1 = prn[1]-p1, pv2 = prn[2]-p2;
            float pv3 = prn[3]-p3, pv4 = prn[4]-p4, pv5 = prn[5]-p5;
            float tv0 = h0-g0, tv1 = h1-g1, tv2 = h2-g2;
            float tv3 = h3-g3, tv4 = h4-g4, tv5 = h5-g5;
            s2 += fabsf(pv0-tv0) + fabsf(pv1-tv1) + fabsf(pv2-tv2);
            s3 += fabsf(pv3-tv3) + fabsf(pv4-tv4) + fabsf(pv5-tv5);

            // quaternion chain
            Quat Q0  = qexp3(g3, g4, g5);
            Quat Q1  = qexp3(h3, h4, h5);
            Quat Q0i = { Q0.w, -Q0.x, -Q0.y, -Q0.z };
            float tq0, tq1, tq2;
            qlog3(qmul(Q0i, Q1), tq0, tq1, tq2);

            // IMU integration
            const float* ir = imu + (base + t) * 10;
            float ts0 = ir[9];
            float ts1 = ir[10 + 9];
            float dt  = (ts1 - ts0) / 1e9f;
            float ax_ = ir[0], ay_ = ir[1], az_ = ir[2];
            float vx_ = ir[3], vy_ = ir[4], vz_ = ir[5];
            float wx_ = ir[6], wy_ = ir[7], wz_ = ir[8];
            float it0 = vx_*dt + 0.5f*ax_*dt*dt;
            float it1 = vy_*dt + 0.5f*ay_*dt*dt;
            float it2 = vz_*dt + 0.5f*az_*dt*dt;
            s4 += fabsf(pv0-it0) + fabsf(pv1-it1) + fabsf(pv2-it2);

            float wn    = fmaxf(sqrtf(wx_*wx_ + wy_*wy_ + wz_*wz_), EPSF);
            float theta = wn * dt;
            float k     = sinf(0.5f*theta) / wn;
            Quat dq = { cosf(0.5f*theta), wx_*k, wy_*k, wz_*k };
            Quat NQ = qmul(Q0, dq);
            float iq0, iq1, iq2;
            qlog3(qmul(Q0i, NQ), iq0, iq1, iq2);
            s5 += fabsf(tq0-iq0) + fabsf(tq1-iq1) + fabsf(tq2-iq2);
        }
    }

    // deterministic wave32 reduction
    #pragma unroll
    for (int off = 16; off > 0; off >>= 1) {
        s0 += __shfl_down(s0, off, 32);
        s1 += __shfl_down(s1, off, 32);
        s2 += __shfl_down(s2, off, 32);
        s3 += __shfl_down(s3, off, 32);
        s4 += __shfl_down(s4, off, 32);
        s5 += __shfl_down(s5, off, 32);
    }
    __shared__ float red[NTHR/32][6];
    const int wave = tid >> 5, lane = tid & 31;
    if (lane == 0) {
        red[wave][0] = s0; red[wave][1] = s1; red[wave][2] = s2;
        red[wave][3] = s3; red[wave][4] = s4; red[wave][5] = s5;
    }
    __syncthreads();
    if (tid == 0) {
        float r[6] = {0,0,0,0,0,0};
        #pragma unroll
        for (int w = 0; w < NTHR/32; ++w)
            #pragma unroll
            for (int j = 0; j < 6; ++j) r[j] += red[w][j];
        float* o = ws + (long long)blockIdx.x * 6;
        #pragma unroll
        for (int j = 0; j < 6; ++j) o[j] = r[j];
    }
}

// Single wave. Lane L serially sums blocks L, L+32, ... (deterministic),
// then each channel is wave-reduced with V_WMMA_F32_16X16X4_F32:
// B = ones so D[m,n] = rowsum(A)[m]; sum of one lane's 8 D regs plus its
// lane^16 partner's 8 regs == sum of all 64 A slots (layout-robust).
__global__ __launch_bounds__(32) void mapnet_finalize(
        const float* __restrict__ ws,
        const float* __restrict__ saxp, const float* __restrict__ saqp,
        const float* __restrict__ srxp, const float* __restrict__ srqp,
        float* __restrict__ out) {
    const int lane = threadIdx.x;
    float p[6] = {0,0,0,0,0,0};
    for (int b = lane; b < NN; b += 32) {
        const float* o = ws + (long long)b * 6;
        #pragma unroll
        for (int j = 0; j < 6; ++j) p[j] += o[j];
    }

    v2f bones; bones[0] = 1.0f; bones[1] = 1.0f;
    float tot[6];
    #pragma unroll
    for (int j = 0; j < 6; ++j) {
        v2f a; a[0] = p[j]; a[1] = 0.0f;
        v8f c = {};
        v8f d = __builtin_amdgcn_wmma_f32_16x16x4_f32(
            /*neg_a=*/false, a, /*neg_b=*/false, bones,
            /*c_mod=*/(short)0, c, /*reuse_a=*/false, /*reuse_b=*/false);
        float ls = ((d[0] + d[1]) + (d[2] + d[3])) + ((d[4] + d[5]) + (d[6] + d[7]));
        ls += __shfl_xor(ls, 16, 32);
        tot[j] = ls;
    }

    if (lane == 0) {
        const float sax = saxp[0], saq = saqp[0], srx = srxp[0], srq = srqp[0];
        const float cntA = 2048.0f * 1024.0f * 3.0f;   // N*T*3
        const float cntV = 2048.0f * 1023.0f * 3.0f;   // N*(T-1)*3
        float m0 = tot[0] / cntA, m1 = tot[1] / cntA;
        float m2 = tot[2] / cntV, m3 = tot[3] / cntV;
        float m4 = tot[4] / cntV, m5 = tot[5] / cntV;
        float abs_loss = expf(-sax)*m0 + sax + expf(-saq)*m1 + saq;
        float vo_loss  = expf(-srx)*m2 + srx + expf(-srq)*m3 + srq;
        float imu_loss = expf(-srx)*m4 + srx + expf(-srq)*m5 + srq;
        out[0] = abs_loss + vo_loss + imu_loss;
    }
}

extern "C" void kernel_launch(void* const* d_in, const int* in_sizes, int n_in,
                              void* d_out, int out_size, void* d_ws, size_t ws_size,
                              hipStream_t stream) {
    const float* pred = (const float*)d_in[0];
    const float* targ = (const float*)d_in[1];
    const float* imu  = (const float*)d_in[2];
    const float* sax  = (const float*)d_in[3];
    const float* saq  = (const float*)d_in[4];
    const float* srx  = (const float*)d_in[5];
    const float* srq  = (const float*)d_in[6];
    float* ws = (float*)d_ws;   // 2048*6 floats = 48 KB of scratch

    mapnet_partial<<<NN, NTHR, 0, stream>>>(pred, targ, imu, ws);
    mapnet_finalize<<<1, 32, 0, stream>>>(ws, sax, saq, srx, srq, (float*)d_out);
}